// etp_l1m1_kl2m2_2_kl3m3_56495999812194
// MI455X (gfx1250) — compile-verified
//
#include <hip/hip_runtime.h>

typedef __attribute__((ext_vector_type(2))) float v2f;
typedef __attribute__((ext_vector_type(8))) float v8f;

#define NPATH  23
#define KCOLS  99
#define KSTEPS 25   // ceil(100/4), col 99 is a zero pad column
#define NCOLS  (KSTEPS * 4)

// ---- compile-time path metadata (LS=[0,1,2,3], MUL=128) -------------------
constexpr int P_L3[NPATH]   = {0,1,2,3, 1,0,2,1,3,2, 2,1,3,0,2,1,3, 3,2,1,3,0,2};
constexpr int P_D1[NPATH]   = {1,1,1,1, 3,3,3,3,3,3, 5,5,5,5,5,5,5, 7,7,7,7,7,7};
constexpr int P_D2[NPATH]   = {1,3,5,7, 1,3,3,5,5,7, 1,3,3,5,5,7,7, 1,3,5,5,7,7};
constexpr int P_D3[NPATH]   = {1,3,5,7, 3,1,5,3,7,5, 5,3,7,1,5,3,7, 7,5,3,7,1,5};
constexpr int P_CG[NPATH]   = {0,1,10,35, 84,93,102,147,192,297,
                               402,427,472,577,602,727,832,
                               1077,1126,1231,1336,1581,1630};
constexpr int P_YOFF[NPATH] = {0,0,0,0, 1,1,1,1,1,1, 4,4,4,4,4,4,4, 9,9,9,9,9,9};
constexpr int P_HROW[NPATH] = {0,1,4,9, 0,1,1,4,4,9, 0,1,1,4,4,9,9, 0,1,4,4,9,9}; // ROW_OFF[l2]

struct ColTab {
    signed char path[NCOLS];   // path index per B/K column (0 for pad; A col is 0)
    signed char hrow[NCOLS];   // h row index (0..15) per column
    signed char b[NCOLS];      // b within the path's d2
    signed char valid[NCOLS];  // 0 for pad columns
};

constexpr ColTab makeCols() {
    ColTab t{};
    int c = 0;
    for (int p = 0; p < NPATH; ++p)
        for (int bb = 0; bb < P_D2[p]; ++bb) {
            t.path[c]  = (signed char)p;
            t.hrow[c]  = (signed char)(P_HROW[p] + bb);
            t.b[c]     = (signed char)bb;
            t.valid[c] = 1;
            ++c;
        }
    for (; c < NCOLS; ++c) {   // pad: A column forced to zero, B value irrelevant
        t.path[c] = 0; t.hrow[c] = 0; t.b[c] = 0; t.valid[c] = 0;
    }
    return t;
}
constexpr ColTab CT = makeCols();

// ---------------------------------------------------------------------------
// One wave32 per edge.
//   A (16x100 f32) built once from y and CGC (block-diagonal per l3):
//       A[row(l3,c), col(p,b)] = sum_a cg_p[a,b,c] * y[a]
//   per u-tile (8): preload hv[16], wv[23] with immediate-offset loads,
//       B[col,u] = hv[hrow(col)] * wv[path(col)] via value-selects (no dynamic
//       array indexing -> no SROA select cascades),
//       C = A @ B with 25 chained v_wmma_f32_16x16x4_f32,
//       store 16x16 tile (full coverage of out[n,:]).
// ---------------------------------------------------------------------------
__global__ __launch_bounds__(256)
void etp_wmma_f32_kernel(const float* __restrict__ Ylm,
                         const float* __restrict__ hklm,
                         const float* __restrict__ weight,
                         const float* __restrict__ CGC,
                         float* __restrict__ out,
                         int n_edges)
{
    const int lane = threadIdx.x & 31;
    const int wave = threadIdx.x >> 5;
    const int n    = blockIdx.x * (blockDim.x >> 5) + wave;
    if (n >= n_edges) return;          // wave-uniform exit

    const int  m    = lane & 15;       // A/C row within 16
    const bool hi   = (lane >> 4) != 0;

    // out-row -> (l3, c): rows [0],[1..3],[4..8],[9..15]
    int row_l3, row_c;
    if (m == 0)      { row_l3 = 0; row_c = 0;     }
    else if (m < 4)  { row_l3 = 1; row_c = m - 1; }
    else if (m < 9)  { row_l3 = 2; row_c = m - 4; }
    else             { row_l3 = 3; row_c = m - 9; }

    const float* Yn = Ylm + (size_t)n * 16;

    // ---- build A: Areg[s] holds A[m, 4s+2*half + {0,1}] (f32 A layout) ----
    v2f Areg[KSTEPS];
    for (int s = 0; s < KSTEPS; ++s) {
        #pragma unroll
        for (int j = 0; j < 2; ++j) {
            const int col = 4 * s + (hi ? 2 : 0) + j;
            float v = 0.0f;
            const int p = CT.path[col];
            if (CT.valid[col] && P_L3[p] == row_l3) {
                const int d1 = P_D1[p], d2 = P_D2[p], d3 = P_D3[p];
                const float* cgp = CGC + P_CG[p] + (int)CT.b[col] * d3 + row_c;
                const float* yp  = Yn + P_YOFF[p];
                float acc = 0.0f;
                for (int a = 0; a < d1; ++a)
                    acc += cgp[a * d2 * d3] * yp[a];   // cg[a,b,c] * y[a]
                v = acc;
            }
            Areg[s][j] = v;
        }
    }

    const float* hn = hklm   + (size_t)n * 2048;
    const float* wn = weight + (size_t)n * 2944;
    float*       on = out    + (size_t)n * 2048;
    const int    ucol  = lane & 15;
    const int    mbase = hi ? 8 : 0;

    #pragma unroll 1
    for (int ut = 0; ut < 8; ++ut) {
        const int u = ut * 16 + ucol;
        const float* hu = hn + u;    // per-lane base; row offsets are immediates
        const float* wu = wn + u;
        float*       ob = on + u + mbase * 128;

        // minimal loads: every h element once, every w element once per edge
        float hv[16];
        #pragma unroll
        for (int r = 0; r < 16; ++r) hv[r] = hu[r * 128];
        float wv[NPATH];
        #pragma unroll
        for (int p = 0; p < NPATH; ++p) wv[p] = wu[p * 128];

        v8f acc = {};
        #pragma unroll
        for (int s = 0; s < KSTEPS; ++s) {
            // all array indices are compile-time constants -> pure SSA values;
            // select acts on fmul results, so no dynamic indexing is created
            const float bx_lo = hv[CT.hrow[4*s + 0]] * wv[CT.path[4*s + 0]];
            const float by_lo = hv[CT.hrow[4*s + 1]] * wv[CT.path[4*s + 1]];
            const float bx_hi = hv[CT.hrow[4*s + 2]] * wv[CT.path[4*s + 2]];
            const float by_hi = hv[CT.hrow[4*s + 3]] * wv[CT.path[4*s + 3]];
            v2f bv;
            bv.x = hi ? bx_hi : bx_lo;
            bv.y = hi ? by_hi : by_lo;
            acc = __builtin_amdgcn_wmma_f32_16x16x4_f32(
                      /*neg_a=*/false, Areg[s],
                      /*neg_b=*/false, bv,
                      /*c_mod=*/(short)0, acc,
                      /*reuse_a=*/false, /*reuse_b=*/false);
        }

        // C layout: VGPR j -> row (j + 8*half), N = lane&15
        #pragma unroll
        for (int j = 0; j < 8; ++j)
            ob[j * 128] = acc[j];
    }
}

extern "C" void kernel_launch(void* const* d_in, const int* in_sizes, int n_in,
                              void* d_out, int out_size, void* d_ws, size_t ws_size,
                              hipStream_t stream)
{
    const float* Ylm    = (const float*)d_in[0];
    const float* hklm   = (const float*)d_in[1];
    const float* weight = (const float*)d_in[2];
    const float* CGC    = (const float*)d_in[3];
    float*       out    = (float*)d_out;

    const int n_edges = in_sizes[0] / 16;      // Y_DIM == 16
    const int waves_per_block = 8;             // 256 threads, wave32
    dim3 block(32 * waves_per_block);
    dim3 grid((n_edges + waves_per_block - 1) / waves_per_block);
    etp_wmma_f32_kernel<<<grid, block, 0, stream>>>(Ylm, hklm, weight, CGC, out, n_edges);
}